// Attention_41343355191890
// MI455X (gfx1250) — compile-verified
//
#include <hip/hip_runtime.h>
#include <hip/hip_bf16.h>
#include <stdint.h>

typedef float v2f __attribute__((ext_vector_type(2)));
typedef float v8f __attribute__((ext_vector_type(8)));
typedef unsigned int v4u __attribute__((ext_vector_type(4)));
typedef int v4i __attribute__((ext_vector_type(4)));
typedef int v8i __attribute__((ext_vector_type(8)));

#define D_MODEL 2048
#define NQ 16
#define NKV 8
#define HDIM 128
#define GRP 2
#define NSEQ 8
#define QLEN 16
#define CTXLEN 8192
#define PAGE_SZ 16
#define PAGES_PER_SEQ 512
#define SM_SCALE 0.08838834764831845f
#define NEG_INF_F (-1e30f)

#if __has_builtin(__builtin_amdgcn_tensor_load_to_lds) && __has_builtin(__builtin_amdgcn_s_wait_tensorcnt)
#define HAVE_TDM 1
#else
#define HAVE_TDM 0
#endif

// D = A(16x4) * B(4x16) + C(16x16), fp32, wave32.
static __device__ __forceinline__ v8f wmma_f32(v2f a, v2f b, v8f c) {
    return __builtin_amdgcn_wmma_f32_16x16x4_f32(
        /*neg_a=*/false, a, /*neg_b=*/false, b,
        /*c_mod=*/(short)0, c, /*reuse_a=*/false, /*reuse_b=*/false);
}

#if HAVE_TDM
// TDM: DMA a 16-row x 128-float fp32 tile (row stride = row_stride_elems
// floats) from global memory into LDS at byte offset lds_off.
static __device__ __forceinline__ void tdm_load_tile_16x128(const float* gsrc,
                                                            unsigned lds_off,
                                                            unsigned row_stride_elems) {
    const unsigned long long ga = (unsigned long long)(uintptr_t)gsrc;
    // D# group 0: count=1 (valid user descriptor), lds_addr, global_addr, type=2.
    v4u g0 = { 1u,
               lds_off,
               (unsigned)ga,
               (unsigned)((ga >> 32) & 0x01FFFFFFu) | (2u << 30) };
    // D# group 1: data_size=2 (4B), tensor_dim0=128, tensor_dim1=16,
    // tile_dim0=128, tile_dim1=16, tensor_dim0_stride=row_stride_elems.
    v8i g1 = { (int)(2u << 16),                        // mask=0, data_size=4B
               (int)((128u & 0xFFFFu) << 16),          // tensor_dim0 lo16
               (int)((128u >> 16) | (16u << 16)),      // tensor_dim0 hi16 | tensor_dim1 lo16
               (int)((16u >> 16) | (128u << 16)),      // tensor_dim1 hi16 | tile_dim0
               (int)(16u),                             // tile_dim1=16, tile_dim2=0
               (int)row_stride_elems,                  // tensor_dim0_stride[31:0]
               0,                                      // stride0 hi | stride1 lo
               0 };
    v4i gz = { 0, 0, 0, 0 };
#if defined(__clang_major__) && (__clang_major__ >= 23)
    v8i gz8 = { 0, 0, 0, 0, 0, 0, 0, 0 };
    __builtin_amdgcn_tensor_load_to_lds(g0, g1, gz, gz, gz8, 0);
#else
    __builtin_amdgcn_tensor_load_to_lds(g0, g1, gz, gz, 0);
#endif
}
#endif

// ---------------------------------------------------------------------------
// Kernel 1: fused QKV projection + RoPE + KV scatter into output caches.
// grid.x = 128 "units" (64 q-pairs, 32 k-pairs, 32 v-pairs); grid.y = 8 row
// tiles of 16 tokens. One wave per (unit,row-tile): two 16x16 WMMA tiles at
// head columns h and h+64 so RoPE pairs are co-resident in a lane.
// ---------------------------------------------------------------------------
__global__ __launch_bounds__(32)
void qkv_rope_scatter(const float* __restrict__ x,
                      const float* __restrict__ wq,
                      const float* __restrict__ wk,
                      const float* __restrict__ wv,
                      const int*   __restrict__ positions,
                      const int*   __restrict__ slot_mapping,
                      float* __restrict__ qws,
                      float* __restrict__ kout,
                      float* __restrict__ vout) {
    const int unit = blockIdx.x;        // 0..127
    const int t0   = blockIdx.y * 16;   // token row tile
    const int lane = threadIdx.x;
    const int lo   = lane & 15;
    const int hi   = lane >> 4;

    int kind, n, ht, wstride;
    const float* w;
    if (unit < 64)      { kind = 0; n = unit >> 2;        ht = unit & 3; w = wq; wstride = NQ  * HDIM; }
    else if (unit < 96) { int u = unit - 64; kind = 1; n = u >> 2; ht = u & 3; w = wk; wstride = NKV * HDIM; }
    else                { int u = unit - 96; kind = 2; n = u >> 2; ht = u & 3; w = wv; wstride = NKV * HDIM; }

    const int h1 = ht * 16 + lo;       // 0..63  (first RoPE half)
    const int h2 = h1 + 64;            // second RoPE half
    const int colbase = n * HDIM;

    v8f acc1 = {};
    v8f acc2 = {};
    const float* xrow = x + (size_t)(t0 + lo) * D_MODEL;
    #pragma unroll 8
    for (int kk = 0; kk < D_MODEL / 4; ++kk) {
        const int kd = kk * 4 + 2 * hi;
        v2f a;  a.x  = xrow[kd];
                a.y  = xrow[kd + 1];
        v2f b1; b1.x = w[(size_t)kd       * wstride + colbase + ht * 16 + lo];
                b1.y = w[(size_t)(kd + 1) * wstride + colbase + ht * 16 + lo];
        v2f b2; b2.x = w[(size_t)kd       * wstride + colbase + ht * 16 + 64 + lo];
                b2.y = w[(size_t)(kd + 1) * wstride + colbase + ht * 16 + 64 + lo];
        acc1 = wmma_f32(a, b1, acc1);
        acc2 = wmma_f32(a, b2, acc2);
    }

    #pragma unroll
    for (int j = 0; j < 8; ++j) {
        const int t = t0 + j + 8 * hi;  // token (C-layout row)
        float x1 = acc1[j];
        float x2 = acc2[j];
        float o1, o2;
        if (kind < 2) {                 // RoPE for q and k
            const float pos  = (float)positions[t];
            // theta^(-h/64) = exp(-h * ln(10000)/64)
            const float freq = __expf(-(float)h1 * (9.210340371976184f / 64.0f));
            float sn, cs;
            __sincosf(pos * freq, &sn, &cs);
            o1 = x1 * cs - x2 * sn;
            o2 = x2 * cs + x1 * sn;
        } else {
            o1 = x1;
            o2 = x2;
        }
        if (kind == 0) {
            qws[(size_t)t * (NQ * HDIM) + colbase + h1] = o1;
            qws[(size_t)t * (NQ * HDIM) + colbase + h2] = o2;
        } else {
            const size_t slot = (size_t)slot_mapping[t];
            float* dst = (kind == 1 ? kout : vout) + slot * (NKV * HDIM) + colbase;
            dst[h1] = o1;
            dst[h2] = o2;
        }
    }
}

// ---------------------------------------------------------------------------
// Kernel 2: paged flash attention with TDM-staged K/V pages.
// grid = (kv_head, seq); 2 waves per block (one per GQA head g). Wave 0 DMAs
// each 16x128 K page and V page into a double-buffered LDS tile with
// tensor_load_to_lds (TENSORcnt), both waves consume it: S = Q*K^T via WMMA,
// online softmax (wave32 shuffles), P staged per-wave in LDS, O += P*V via
// WMMA. Output normalized and written directly (no split-K merge needed).
// ---------------------------------------------------------------------------
__global__ __launch_bounds__(64)
void paged_attention(const float* __restrict__ qws,
                     const float* __restrict__ kcache,
                     const float* __restrict__ vcache,
                     const int*   __restrict__ block_tables,
                     const int*   __restrict__ seq_lens,
                     float* __restrict__ attws) {
    __shared__ float kbuf[2][16][HDIM];   // double-buffered K page
    __shared__ float vbuf[2][16][HDIM];   // double-buffered V page
    __shared__ float pbuf[2][16][16];     // per-wave P staging (C->A layout)

    const int kvh  = blockIdx.x;          // 0..7
    const int seq  = blockIdx.y;          // 0..7
    const int wid  = threadIdx.x >> 5;    // g = wid
    const int lane = threadIdx.x & 31;
    const int lo   = lane & 15;
    const int hi   = lane >> 4;

    const int n       = kvh * GRP + wid;  // q head
    const int kvlen   = seq_lens[seq];
    const int nblocks = (kvlen + PAGE_SZ - 1) / PAGE_SZ;

    // Q tile resident in registers as 32 fp32 A-fragments.
    v2f qf[32];
    const float* qrow = qws + (size_t)(seq * QLEN + lo) * (NQ * HDIM) + (size_t)n * HDIM;
    #pragma unroll
    for (int kk = 0; kk < 32; ++kk) {
        qf[kk].x = qrow[kk * 4 + 2 * hi];
        qf[kk].y = qrow[kk * 4 + 2 * hi + 1];
    }

    v8f oacc[8];
    #pragma unroll
    for (int t = 0; t < 8; ++t) oacc[t] = (v8f){};
    float mrow[8], lrow[8];
    #pragma unroll
    for (int j = 0; j < 8; ++j) { mrow[j] = NEG_INF_F; lrow[j] = 0.0f; }

    const int* bt = block_tables + (size_t)seq * PAGES_PER_SEQ;

    // Stage one K/V page (16 keys) into LDS buffer `buf`.
    auto stage_page = [&](int blk, int buf) {
        const int    page   = bt[blk];
        const size_t kvbase = (size_t)page * PAGE_SZ * (NKV * HDIM) + (size_t)kvh * HDIM;
#if HAVE_TDM
        tdm_load_tile_16x128(kcache + kvbase, (unsigned)(uintptr_t)&kbuf[buf][0][0], NKV * HDIM);
        tdm_load_tile_16x128(vcache + kvbase, (unsigned)(uintptr_t)&vbuf[buf][0][0], NKV * HDIM);
#else
        // Fallback: cooperative vector loads by wave 0.
        for (int r = 0; r < 16; ++r) {
            const float4* ks = (const float4*)(kcache + kvbase + (size_t)r * (NKV * HDIM));
            const float4* vs = (const float4*)(vcache + kvbase + (size_t)r * (NKV * HDIM));
            ((float4*)&kbuf[buf][r][0])[lane] = ks[lane];
            ((float4*)&vbuf[buf][r][0])[lane] = vs[lane];
        }
#endif
    };

    if (wid == 0) stage_page(0, 0);

    for (int blk = 0; blk < nblocks; ++blk) {
        const int cur = blk & 1;
        if (wid == 0) {
            if (blk + 1 < nblocks) {
                stage_page(blk + 1, cur ^ 1);
#if HAVE_TDM
                __builtin_amdgcn_s_wait_tensorcnt(2);   // current page complete
#endif
            } else {
#if HAVE_TDM
                __builtin_amdgcn_s_wait_tensorcnt(0);
#endif
            }
        }
        __syncthreads();                  // publish staged page to both waves

        // S = Q * K^T  (16 q-rows x 16 keys, inner dim H=128), K from LDS.
        v8f s = {};
        const float* krow = &kbuf[cur][lo][0];
        #pragma unroll
        for (int kk = 0; kk < 32; ++kk) {
            v2f b; b.x = krow[kk * 4 + 2 * hi];
                   b.y = krow[kk * 4 + 2 * hi + 1];
            s = wmma_f32(qf[kk], b, s);
        }

        // Online softmax (row stats via wave32 xor-shuffles over 16-lane halves).
        const int kb0 = blk * PAGE_SZ;
        float alpha[8];
        #pragma unroll
        for (int j = 0; j < 8; ++j) {
            const int m    = j + 8 * hi;          // local q row
            const int key  = kb0 + lo;
            const int qpos = kvlen - QLEN + m;
            float val = s[j] * SM_SCALE;
            if (key > qpos || key >= kvlen) val = NEG_INF_F;

            float rmax = val;
            rmax = fmaxf(rmax, __shfl_xor(rmax, 1, 32));
            rmax = fmaxf(rmax, __shfl_xor(rmax, 2, 32));
            rmax = fmaxf(rmax, __shfl_xor(rmax, 4, 32));
            rmax = fmaxf(rmax, __shfl_xor(rmax, 8, 32));
            const float mnew = fmaxf(mrow[j], rmax);
            const float a    = __expf(mrow[j] - mnew);
            float p = (val > -1e29f) ? __expf(val - mnew) : 0.0f;
            float rsum = p;
            rsum += __shfl_xor(rsum, 1, 32);
            rsum += __shfl_xor(rsum, 2, 32);
            rsum += __shfl_xor(rsum, 4, 32);
            rsum += __shfl_xor(rsum, 8, 32);
            lrow[j]  = lrow[j] * a + rsum;
            mrow[j]  = mnew;
            alpha[j] = a;
            pbuf[wid][m][lo] = p;                 // C-layout -> LDS
        }
        #pragma unroll
        for (int t = 0; t < 8; ++t) {
            #pragma unroll
            for (int j = 0; j < 8; ++j) oacc[t][j] *= alpha[j];
        }

        // O += P * V   (A = P from LDS, B = V page rows from LDS, 8 dim-tiles)
        #pragma unroll
        for (int kk = 0; kk < 4; ++kk) {
            v2f a;
            a.x = pbuf[wid][lo][kk * 4 + 2 * hi];
            a.y = pbuf[wid][lo][kk * 4 + 2 * hi + 1];
            const float* vrow0 = &vbuf[cur][kk * 4 + 2 * hi][0];
            const float* vrow1 = vrow0 + HDIM;
            #pragma unroll
            for (int t = 0; t < 8; ++t) {
                v2f b; b.x = vrow0[t * 16 + lo];
                       b.y = vrow1[t * 16 + lo];
                oacc[t] = wmma_f32(a, b, oacc[t]);
            }
        }
        __syncthreads();                  // page fully consumed; safe to refill
    }

    // Normalize and write attention output: att[t][n][h].
    #pragma unroll
    for (int j = 0; j < 8; ++j) {
        const int   m    = j + 8 * hi;
        const float invL = 1.0f / lrow[j];
        float* dst = attws + (size_t)(seq * QLEN + m) * (NQ * HDIM) + (size_t)n * HDIM;
        #pragma unroll
        for (int t = 0; t < 8; ++t)
            dst[t * 16 + lo] = oacc[t][j] * invL;
    }
}

// ---------------------------------------------------------------------------
// Kernel 3: output projection  o[t,d] = sum_{n,h} att[t,n,h] * wo[n,h,d].
// One wave per 16x16 output tile, inner dim N*H = 2048.
// ---------------------------------------------------------------------------
__global__ __launch_bounds__(32)
void out_proj(const float* __restrict__ attws,
              const float* __restrict__ wo,
              float* __restrict__ o) {
    const int col0 = blockIdx.x * 16;
    const int t0   = blockIdx.y * 16;
    const int lane = threadIdx.x;
    const int lo   = lane & 15;
    const int hi   = lane >> 4;

    v8f acc = {};
    const float* arow = attws + (size_t)(t0 + lo) * (NQ * HDIM);
    #pragma unroll 8
    for (int kk = 0; kk < (NQ * HDIM) / 4; ++kk) {
        const int kd = kk * 4 + 2 * hi;
        v2f a; a.x = arow[kd];
               a.y = arow[kd + 1];
        v2f b; b.x = wo[(size_t)kd       * D_MODEL + col0 + lo];
               b.y = wo[(size_t)(kd + 1) * D_MODEL + col0 + lo];
        acc = wmma_f32(a, b, acc);
    }
    #pragma unroll
    for (int j = 0; j < 8; ++j)
        o[(size_t)(t0 + j + 8 * hi) * D_MODEL + col0 + lo] = acc[j];
}

extern "C" void kernel_launch(void* const* d_in, const int* in_sizes, int n_in,
                              void* d_out, int out_size, void* d_ws, size_t ws_size,
                              hipStream_t stream) {
    (void)in_sizes; (void)n_in; (void)out_size; (void)ws_size;
    const float* x            = (const float*)d_in[0];
    const float* wq           = (const float*)d_in[1];
    const float* wk           = (const float*)d_in[2];
    const float* wv           = (const float*)d_in[3];
    const float* wo           = (const float*)d_in[4];
    const float* k_cache      = (const float*)d_in[5];
    const float* v_cache      = (const float*)d_in[6];
    const int*   positions    = (const int*)d_in[7];
    const int*   slot_mapping = (const int*)d_in[8];
    const int*   block_tables = (const int*)d_in[9];
    const int*   seq_lens     = (const int*)d_in[10];

    const size_t CACHE_ELEMS = (size_t)4096 * PAGE_SZ * NKV * HDIM;  // 67108864
    float* kout = (float*)d_out;
    float* vout = kout + CACHE_ELEMS;
    float* oout = vout + CACHE_ELEMS;

    float* qws   = (float*)d_ws;                       // [128][16][128]
    float* attws = qws + (size_t)128 * NQ * HDIM;      // [128][16][128]

    // new_cache = old_cache with updated slots: bulk copy, then scatter.
    hipMemcpyAsync(kout, k_cache, CACHE_ELEMS * sizeof(float), hipMemcpyDeviceToDevice, stream);
    hipMemcpyAsync(vout, v_cache, CACHE_ELEMS * sizeof(float), hipMemcpyDeviceToDevice, stream);

    qkv_rope_scatter<<<dim3(128, 8), 32, 0, stream>>>(x, wq, wk, wv, positions,
                                                      slot_mapping, qws, kout, vout);
    paged_attention<<<dim3(NKV, NSEQ), 64, 0, stream>>>(qws, kout, vout,
                                                        block_tables, seq_lens, attws);
    out_proj<<<dim3(D_MODEL / 16, 8), 32, 0, stream>>>(attws, wo, oout);
}